// ConvolutionalLayer_64338610094108
// MI455X (gfx1250) — compile-verified
//
#include <hip/hip_runtime.h>

typedef __attribute__((ext_vector_type(2))) float v2f;
typedef __attribute__((ext_vector_type(8))) float v8f;

#define IMG_H 512
#define IMG_W 512
#define CIN   16
#define COUT  64
#define THRESH 1.0f

// One wave computes a 16-pixel (M) x 64-channel (N) tile of the implicit GEMM:
//   out[p, co] = sum_{ky,kx,ci} in[y+ky-1, x+kx-1, ci] * w[ky,kx,ci,co]
// K = 3*3*16 = 144, via 144 x V_WMMA_F32_16X16X4_F32.
//
// Weights live in LDS pre-swizzled into B-fragment order (pair of K rows
// contiguous per lane) so one ds_load_b64 = one B fragment and the compiler
// can merge pairs into ds_load_2addr_b64.
//
// All 36 A fragments are loaded into registers up front (one big VMEM
// clause), and B fragments are software-pipelined one c-step ahead, so the
// WMMA stream never stalls on loadcnt and tolerates dscnt >= 1.
__global__ __launch_bounds__(256) void snn_conv3x3_wmma_f32(
    const float* __restrict__ in,    // [512][512][16]
    const float* __restrict__ wt,    // [3][3][16][64]
    const float* __restrict__ oldp,  // [512][512][64]
    float* __restrict__ spikes,      // [512][512][64]
    float* __restrict__ newp)        // [512][512][64]
{
    __shared__ float lw[9 * 8 * COUT * 2];   // 36 KB, pair-swizzled weights

    const int tid = threadIdx.x;
    for (int i = tid; i < 9 * CIN * COUT; i += 256) {
        const int co = i & 63;
        const int ci = (i >> 6) & 15;
        const int kk = i >> 10;
        lw[(((kk << 3) + (ci >> 1)) * COUT + co) * 2 + (ci & 1)] = wt[i];
    }
    __syncthreads();

    const int lane = tid & 31;
    const int wave = tid >> 5;
    const int m    = lane & 15;     // M row (pixel) / N col (channel) within tile
    const int half = lane >> 4;     // 0: lanes 0-15, 1: lanes 16-31

    const int y  = blockIdx.x >> 2;                          // image row
    const int x0 = ((blockIdx.x & 3) << 7) + (wave << 4);    // strip start column

    const int koff = half << 1;   // A layout: lanes 0-15 carry K={0,1}, 16-31 K={2,3}

    // ---- Phase 1: load ALL 36 A fragments (9 taps x 4 c-steps) ----
    v2f a[9][4];
    #pragma unroll
    for (int ky = 0; ky < 3; ++ky) {
        const int gy  = y + ky - 1;
        const int gyc = gy < 0 ? 0 : (gy > IMG_H - 1 ? IMG_H - 1 : gy);
        #pragma unroll
        for (int kx = 0; kx < 3; ++kx) {
            const int gx  = x0 + m + kx - 1;
            const int gxc = gx < 0 ? 0 : (gx > IMG_W - 1 ? IMG_W - 1 : gx);
            const float* ap = in + ((size_t)(gyc * IMG_W + gxc) * CIN) + koff;
            #pragma unroll
            for (int cs = 0; cs < 4; ++cs)
                a[ky * 3 + kx][cs] = *(const v2f*)(ap + 4 * cs);
        }
    }
    // Apply SAME-padding masks (cndmask; loads above stay unconditional).
    #pragma unroll
    for (int ky = 0; ky < 3; ++ky) {
        const int  gy    = y + ky - 1;
        const bool rowok = ((unsigned)gy < (unsigned)IMG_H);
        #pragma unroll
        for (int kx = 0; kx < 3; ++kx) {
            const int  gx = x0 + m + kx - 1;
            const bool ok = rowok && ((unsigned)gx < (unsigned)IMG_W);
            #pragma unroll
            for (int cs = 0; cs < 4; ++cs) {
                a[ky * 3 + kx][cs].x = ok ? a[ky * 3 + kx][cs].x : 0.f;
                a[ky * 3 + kx][cs].y = ok ? a[ky * 3 + kx][cs].y : 0.f;
            }
        }
    }

    // ---- Phase 2: WMMA stream with B software-pipelined from LDS ----
    const v8f zero8 = {0.f, 0.f, 0.f, 0.f, 0.f, 0.f, 0.f, 0.f};
    v8f acc[4];
    acc[0] = zero8; acc[1] = zero8; acc[2] = zero8; acc[3] = zero8;

    // B fragment address: pair row ((kk*8) + (c>>1) + half), col n*16+m,
    // 2 floats per (pair,col).
    const float* bwave = lw + ((size_t)((half * COUT) + m) << 1);
#define FETCH_B(dst, kk_, cs_)                                                  \
    {                                                                           \
        const float* bp_ = bwave + (size_t)(((kk_) << 3) + ((cs_) << 1)) *      \
                                       (COUT * 2);                              \
        _Pragma("unroll")                                                       \
        for (int n_ = 0; n_ < 4; ++n_)                                          \
            dst[n_] = *(const v2f*)(bp_ + ((n_ << 4) << 1));                    \
    }

    v2f bcur[4], bnxt[4];
    FETCH_B(bcur, 0, 0)

    #pragma unroll
    for (int kk = 0; kk < 9; ++kk) {
        #pragma unroll
        for (int cs = 0; cs < 4; ++cs) {
            // Prefetch next step's B fragments before consuming current ones.
            const int kk2 = (cs == 3) ? kk + 1 : kk;
            const int cs2 = (cs == 3) ? 0 : cs + 1;
            if (kk2 < 9) FETCH_B(bnxt, kk2, cs2)
            #pragma unroll
            for (int n = 0; n < 4; ++n)
                acc[n] = __builtin_amdgcn_wmma_f32_16x16x4_f32(
                    false, a[kk][cs], false, bcur[n], (short)0, acc[n],
                    false, false);
            #pragma unroll
            for (int n = 0; n < 4; ++n) bcur[n] = bnxt[n];
        }
    }
#undef FETCH_B

    // ---- Epilogue: C/D layout VGPR r -> (M = r + 8*half, N = lane%16). ----
    const int pbase = y * IMG_W + x0 + (half << 3);
    #pragma unroll
    for (int n = 0; n < 4; ++n) {
        const int ch = (n << 4) + m;
        #pragma unroll
        for (int r = 0; r < 8; ++r) {
            const size_t idx = (size_t)(pbase + r) * COUT + ch;
            float v = acc[n][r] + oldp[idx];
            const bool s = (v >= THRESH);
            spikes[idx] = s ? 1.0f : 0.0f;
            newp[idx]   = s ? 0.0f : v;
        }
    }
}

extern "C" void kernel_launch(void* const* d_in, const int* in_sizes, int n_in,
                              void* d_out, int out_size, void* d_ws, size_t ws_size,
                              hipStream_t stream) {
    const float* in  = (const float*)d_in[0];   // input_to_layer [1,512,512,16]
    const float* wt  = (const float*)d_in[1];   // weights        [3,3,16,64]
    const float* op  = (const float*)d_in[2];   // old_potentials [1,512,512,64]
    float* spikes = (float*)d_out;                              // output 0
    float* newp   = spikes + (size_t)IMG_H * IMG_W * COUT;      // output 1

    // 512 rows * 4 blocks/row (128 px each), 256 threads = 8 waves of 16 px
    dim3 grid(IMG_H * (IMG_W / 128));
    snn_conv3x3_wmma_f32<<<grid, 256, 0, stream>>>(in, wt, op, spikes, newp);
}